// ContraAttention_73478300500658
// MI455X (gfx1250) — compile-verified
//
#include <hip/hip_runtime.h>
#include <hip/hip_bf16.h>
#include <math.h>

// Problem constants (from reference)
#define NB   128   // batch
#define LQ   64
#define LK   64
#define DD   512   // model dim (h*dh folded back together)

typedef __attribute__((ext_vector_type(2))) float v2f;
typedef __attribute__((ext_vector_type(8))) float v8f;

// ---------------------------------------------------------------------------
// V_WMMA_F32_16X16X4_F32 operand addressing from a row-major matrix (stride DD):
//   A (16x4 MxK) and B (4x16 KxN, built as K-tile^T) both reduce to:
//   lane loads float2 at row (lane&15), cols kb + (lane>>4)*2.
// C/D tile layout: acc[v], lanes 0-15 -> row v, col=lane; lanes 16-31 -> row
//   v+8, col=lane-16.
// ---------------------------------------------------------------------------

__device__ __forceinline__ float butterfly_sum16(float x) {  // within 16-lane half
    x += __shfl_xor(x, 1, 32);
    x += __shfl_xor(x, 2, 32);
    x += __shfl_xor(x, 4, 32);
    x += __shfl_xor(x, 8, 32);
    return x;
}
__device__ __forceinline__ float butterfly_sum32(float x) {
    return butterfly_sum16(x) + __shfl_xor(butterfly_sum16(x), 16, 32);
}

// ===========================================================================
// Kernel 1: fused projection  out = X @ W^T + b  (all fp32, row-major)
//   X: (8192,512)  W: (512,512)  out: (8192,512)
//   2x2 register blocking: one wave owns a 32x32 output block.
//   blockIdx.z == 0 -> Q projection, == 1 -> K projection.
// ===========================================================================
__global__ __launch_bounds__(128)
void ca_proj_kernel(const float* __restrict__ q_in, const float* __restrict__ k_in,
                    const float* __restrict__ Wq, const float* __restrict__ bq,
                    const float* __restrict__ Wk, const float* __restrict__ bk,
                    float* __restrict__ Qp, float* __restrict__ Kp) {
    const bool isK = (blockIdx.z != 0);
    const float* __restrict__ X    = isK ? k_in : q_in;
    const float* __restrict__ W    = isK ? Wk   : Wq;
    const float* __restrict__ bias = isK ? bk   : bq;
    float* __restrict__ out        = isK ? Kp   : Qp;

    const int lane = threadIdx.x & 31;
    const int wave = threadIdx.x >> 5;
    const int tile = blockIdx.x * 4 + wave;        // 0 .. 4095 (256 x 16 blocks)
    const int rowbase = (tile >> 4) * 32;
    const int colbase = (tile & 15) * 32;

    const int r    = lane & 15;
    const int koff = (lane >> 4) * 2;

    const float* __restrict__ Abase = X + (size_t)(rowbase + r) * DD + koff;
    const float* __restrict__ Bbase = W + (size_t)(colbase + r) * DD + koff;

    v8f acc[2][2] = {};
#pragma unroll 2
    for (int kb = 0; kb < DD; kb += 4) {
        v2f a0 = *(const v2f*)(Abase + kb);
        v2f a1 = *(const v2f*)(Abase + 16 * DD + kb);   // +32KB imm offset
        v2f b0 = *(const v2f*)(Bbase + kb);
        v2f b1 = *(const v2f*)(Bbase + 16 * DD + kb);
        acc[0][0] = __builtin_amdgcn_wmma_f32_16x16x4_f32(false, a0, false, b0, (short)0, acc[0][0], false, false);
        acc[0][1] = __builtin_amdgcn_wmma_f32_16x16x4_f32(false, a0, false, b1, (short)0, acc[0][1], false, false);
        acc[1][0] = __builtin_amdgcn_wmma_f32_16x16x4_f32(false, a1, false, b0, (short)0, acc[1][0], false, false);
        acc[1][1] = __builtin_amdgcn_wmma_f32_16x16x4_f32(false, a1, false, b1, (short)0, acc[1][1], false, false);
    }

    const float bv0 = bias[colbase + r];
    const float bv1 = bias[colbase + 16 + r];
    const int   rhi = (lane >> 4) << 3;   // +8 rows for the upper half-wave
#pragma unroll
    for (int i = 0; i < 2; ++i)
#pragma unroll
        for (int v = 0; v < 8; ++v) {
            const size_t row = (size_t)(rowbase + i * 16 + v + rhi) * DD;
            out[row + colbase + r]      = acc[i][0][v] + bv0;
            out[row + colbase + 16 + r] = acc[i][1][v] + bv1;
        }
}

// ===========================================================================
// Kernel 2: one WAVE per (a,b) pair.
//   S = Qp[a] @ Kp[b]^T (64x64) as a 4x4 grid of WMMA tiles (128 C VGPRs).
//   16 independent accumulation chains; 8 float2 loads feed 16 WMMAs per
//   k-step (16 FLOP/B).  All max/sum reductions stay in-wave via shuffles:
//   no LDS, no barriers.  Never materializes the (N,N,64,64) logits tensor.
// ===========================================================================
__global__ __launch_bounds__(128)
void ca_pair_kernel(const float* __restrict__ Qp, const float* __restrict__ Kp,
                    const float* __restrict__ attn_mask,
                    const float* __restrict__ logit_scale,
                    float* __restrict__ out) {
    const int lane = threadIdx.x & 31;
    const int wave = threadIdx.x >> 5;
    const int pair = blockIdx.x * 4 + wave;   // 0 .. 16383
    const int a = pair >> 7;
    const int b = pair & (NB - 1);

    const int r    = lane & 15;
    const int koff = (lane >> 4) * 2;

    const float* __restrict__ Abase = Qp + (size_t)(a * LQ + r) * DD + koff;
    const float* __restrict__ Bbase = Kp + (size_t)(b * LK + r) * DD + koff;

    v8f acc[4][4] = {};
#pragma unroll 1
    for (int kb = 0; kb < DD; kb += 4) {
        v2f av[4], bv[4];
#pragma unroll
        for (int t = 0; t < 4; ++t) {
            av[t] = *(const v2f*)(Abase + (size_t)t * 16 * DD + kb);  // +t*32KB
            bv[t] = *(const v2f*)(Bbase + (size_t)t * 16 * DD + kb);
        }
#pragma unroll
        for (int ti = 0; ti < 4; ++ti)
#pragma unroll
            for (int tj = 0; tj < 4; ++tj)
                acc[ti][tj] = __builtin_amdgcn_wmma_f32_16x16x4_f32(
                    false, av[ti], false, bv[tj], (short)0, acc[ti][tj], false, false);
    }

    // ---- t2v: row maxima over all 64 columns, then sum over rows ----------
    // Row 16ti+v(+8*half): max over tj elementwise, then 4-step butterfly max
    // within each 16-lane half (xor<16 never crosses halves).
    float thalf = 0.0f;
#pragma unroll
    for (int ti = 0; ti < 4; ++ti)
#pragma unroll
        for (int v = 0; v < 8; ++v) {
            float m = fmaxf(fmaxf(acc[ti][0][v], acc[ti][1][v]),
                            fmaxf(acc[ti][2][v], acc[ti][3][v]));
            m = fmaxf(m, __shfl_xor(m, 1, 32));
            m = fmaxf(m, __shfl_xor(m, 2, 32));
            m = fmaxf(m, __shfl_xor(m, 4, 32));
            m = fmaxf(m, __shfl_xor(m, 8, 32));
            thalf += m;   // low half: rows 16ti+v ; high half: rows 16ti+v+8
        }
    const float tsum = thalf + __shfl_xor(thalf, 16, 32);  // all 64 rows

    // ---- v2t: column maxima over all 64 rows, then sum over columns -------
    float csum_lane = 0.0f;
#pragma unroll
    for (int tj = 0; tj < 4; ++tj) {
        float c = acc[0][tj][0];
#pragma unroll
        for (int ti = 0; ti < 4; ++ti)
#pragma unroll
            for (int v = 0; v < 8; ++v)
                c = fmaxf(c, acc[ti][tj][v]);
        c = fmaxf(c, __shfl_xor(c, 16, 32));  // combine row halves
        csum_lane += c;   // lane holds colmax for col tj*16 + (lane&15)
    }
    const float csum = butterfly_sum16(csum_lane);  // sum of 16 distinct cols x4 tj

    // ---- com_mask_sum[a] ---------------------------------------------------
    const float msum = butterfly_sum32(attn_mask[a * LQ + lane] +
                                       attn_mask[a * LQ + lane + 32]);

    if (lane == 0) {
        const float t2v = tsum / msum;
        const float v2t = csum / (float)LK;
        const float rv  = 0.5f * (t2v + v2t) * expf(logit_scale[0]);
        out[a * NB + b] = rv;                 // r
        out[NB * NB + b * NB + a] = rv;       // r.T
    }
}

// ===========================================================================
extern "C" void kernel_launch(void* const* d_in, const int* in_sizes, int n_in,
                              void* d_out, int out_size, void* d_ws, size_t ws_size,
                              hipStream_t stream) {
    const float* q    = (const float*)d_in[0];   // (128,64,512)
    const float* k    = (const float*)d_in[1];   // (128,64,512)
    const float* mask = (const float*)d_in[2];   // (128,64)
    const float* Wq   = (const float*)d_in[3];   // (512,512)
    const float* bq   = (const float*)d_in[4];   // (512,)
    const float* Wk   = (const float*)d_in[5];   // (512,512)
    const float* bk   = (const float*)d_in[6];   // (512,)
    const float* ls   = (const float*)d_in[7];   // scalar
    float* out = (float*)d_out;

    float* Qp = (float*)d_ws;                    // 8192x512 fp32 = 16 MB
    float* Kp = Qp + (size_t)NB * LQ * DD;       // 8192x512 fp32 = 16 MB

    // Stage 1: Q/K projections. 4096 wave-tiles (32x32 each) per projection.
    ca_proj_kernel<<<dim3(1024, 1, 2), 128, 0, stream>>>(q, k, Wq, bq, Wk, bk, Qp, Kp);

    // Stage 2: all-pairs scores + fused reductions, one wave per pair.
    ca_pair_kernel<<<dim3(4096), 128, 0, stream>>>(Qp, Kp, mask, ls, out);
}